// ChirpTextureSynth_1116691497161
// MI455X (gfx1250) — compile-verified
//
#include <hip/hip_runtime.h>
#include <math.h>

// ---------------- problem constants ----------------
#define SRf        44100.0f
#define NSAMP      524288          // 2^19 output samples
#define NGRAIN     4096
#define GRAINN     16384
#define SEG        1024            // output samples per block in k_synth
#define NBLK_MAIN  (NSAMP / SEG)   // 512 blocks
#define TPB        256             // 8 wave32 per block

// ---------------- workspace layout (bytes) ----------------
// [0      , 16384)  : onsets copy  (int[4096])
// [16384  , 32768)  : f0 copy      (float[4096])
// [32768  , 49152)  : scale        (float[4096])      <- 48KB contiguous "descriptor block"
// [49152  , 180224) : {phase,win}  (float2[16384])
// [180224 , 180228) : norm accumulator (float)
#define WS_ON    0
#define WS_F0    16384
#define WS_SC    32768
#define WS_DESC_BYTES 49152
#define WS_PW    49152
#define WS_NORM  (49152 + 131072)

// ============ kernel 0: per-grain scales + zero norm ============
__global__ __launch_bounds__(TPB)
void k_prep_grains(const float* __restrict__ p_dens,
                   const float* __restrict__ f0,
                   const int*   __restrict__ onsets,
                   int*   __restrict__ ws_on,
                   float* __restrict__ ws_f0,
                   float* __restrict__ ws_sc,
                   float* __restrict__ ws_norm) {
    __shared__ float red[TPB];
    const int t = threadIdx.x;
    const float d   = p_dens[0];
    const float off = 0.25f * d + 0.75f * d * d;
    const float k1  = (1.0f - d) * (float)NGRAIN;

    float amp[16];
    float m = -1.0f;
#pragma unroll
    for (int i = 0; i < 16; ++i) {
        const int g = t * 16 + i;
        const float sig_op = k1 * ((float)g * (1.0f / (float)NGRAIN) - off);
        const float sig = 1.0f / (1.0f + expf(-2.0f * sig_op));
        const float a = 1.0f - sig;
        amp[i] = a;
        m = fmaxf(m, a);
    }
    red[t] = m;
    __syncthreads();
    for (int s = TPB / 2; s > 0; s >>= 1) {
        if (t < s) red[t] = fmaxf(red[t], red[t + s]);
        __syncthreads();
    }
    const float maxamp = red[0];
#pragma unroll
    for (int i = 0; i < 16; ++i) {
        const int g = t * 16 + i;
        const float f = f0[g];
        ws_on[g] = onsets[g];
        ws_f0[g] = f;
        ws_sc[g] = amp[i] / (maxamp * sqrtf(f));
    }
    if (t == 0) ws_norm[0] = 0.0f;
}

// ============ kernel 1: phase + hann window table ============
__global__ __launch_bounds__(TPB)
void k_prep_pw(const float* __restrict__ p_slope, float2* __restrict__ pw) {
    const int l = blockIdx.x * TPB + threadIdx.x;
    const float ts      = p_slope[0];
    const float typical = SRf / (float)(12 * 256);             // SR/(Q*HOP_LEN)
    const float gamma   = tanf(ts * 1.57079632679489662f) * typical * 0.25f;
    const float tt      = (float)l * (1.0f / SRf) - ((float)GRAINN / SRf) * 0.5f;
    float phase;
    if (gamma == 0.0f) phase = tt;
    else               phase = expm1f(gamma * tt) / gamma;     // CONST_LOG2 == 1
    const float ws = sinf(3.14159265358979323846f * (float)l * (1.0f / (float)GRAINN));
    pw[l] = make_float2(phase, ws * ws);
}

// ============ kernel 2: gather synthesis (no output atomics) ============
__global__ __launch_bounds__(TPB)
void k_synth(const void*  __restrict__ ws_desc,   // 48KB {onset,f0,scale}
             const float2* __restrict__ pw,
             float* __restrict__ out,
             float* __restrict__ ws_norm) {
    __shared__ int   s_desc[WS_DESC_BYTES / 4];   // 48KB
    __shared__ float s_wred[TPB / 32];

    const int t = threadIdx.x;

    // --- async stage of descriptor block into LDS (CDNA5 async path) ---
    {
        const unsigned lds_base = (unsigned)(size_t)(&s_desc[0]); // low 32b of flat = LDS offset
#pragma unroll
        for (int it = 0; it < 12; ++it) {                 // 12 * 256 * 16B = 48KB
            const unsigned go = (unsigned)((it * TPB + t) * 16);
            const unsigned lo = lds_base + go;
            asm volatile("global_load_async_to_lds_b128 %0, %1, %2"
                         :: "v"(lo), "v"(go), "s"(ws_desc) : "memory");
        }
        asm volatile("s_wait_asynccnt 0x0" ::: "memory");
    }
    __syncthreads();

    const int*   son = s_desc;
    const float* sf0 = (const float*)&s_desc[NGRAIN];
    const float* ssc = (const float*)&s_desc[2 * NGRAIN];

    const int s0 = blockIdx.x * SEG;
    float acc[4] = {0.0f, 0.0f, 0.0f, 0.0f};

    for (int g = 0; g < NGRAIN; ++g) {
        const int o = son[g];                                  // uniform across block
        if (o >= s0 + SEG || o + GRAINN <= s0) continue;       // uniform branch
        const float f  = sf0[g];
        const float sc = ssc[g];
#pragma unroll
        for (int k = 0; k < 4; ++k) {
            const int n = s0 + k * TPB + t;
            const unsigned l = (unsigned)(n - o);
            if (l < (unsigned)GRAINN) {
                const float2 p = pw[l];                        // coalesced, L2-resident
                const float arg = f * p.x;                     // revolutions (cycles)
                const float r = arg - rintf(arg);              // |r| <= 0.5
                const float s = __builtin_amdgcn_sinf(r);      // v_sin_f32: sin(2*pi*r)
                acc[k] = fmaf(s * p.y, sc, acc[k]);
            }
        }
    }

    // store (each output sample written exactly once) + fused sum-of-squares
    float sq = 0.0f;
#pragma unroll
    for (int k = 0; k < 4; ++k) {
        out[s0 + k * TPB + t] = acc[k];
        sq = fmaf(acc[k], acc[k], sq);
    }
    // wave32 reduction
    for (int o2 = 16; o2 > 0; o2 >>= 1) sq += __shfl_down(sq, o2, 32);
    if ((t & 31) == 0) s_wred[t >> 5] = sq;
    __syncthreads();
    if (t == 0) {
        float tot = 0.0f;
#pragma unroll
        for (int i = 0; i < TPB / 32; ++i) tot += s_wred[i];
        atomicAdd(ws_norm, tot);                               // 512 atomics total
    }
}

// ============ kernel 3: normalize ============
__global__ __launch_bounds__(TPB)
void k_norm(float* __restrict__ out, const float* __restrict__ ws_norm) {
    const int i = blockIdx.x * TPB + threadIdx.x;
    const float inv = 1.0f / sqrtf(ws_norm[0]);
    out[i] *= inv;
}

// ============ entry point ============
extern "C" void kernel_launch(void* const* d_in, const int* in_sizes, int n_in,
                              void* d_out, int out_size, void* d_ws, size_t ws_size,
                              hipStream_t stream) {
    const float* dens   = (const float*)d_in[0];   // theta_density (scalar)
    const float* slope  = (const float*)d_in[1];   // theta_slope   (scalar)
    const float* f0     = (const float*)d_in[2];   // [4096]
    const int*   onsets = (const int*)  d_in[3];   // [4096]

    char* ws = (char*)d_ws;
    int*    ws_on   = (int*)   (ws + WS_ON);
    float*  ws_f0   = (float*) (ws + WS_F0);
    float*  ws_sc   = (float*) (ws + WS_SC);
    float2* pw      = (float2*)(ws + WS_PW);
    float*  ws_norm = (float*) (ws + WS_NORM);
    float*  out     = (float*) d_out;

    hipLaunchKernelGGL(k_prep_grains, dim3(1), dim3(TPB), 0, stream,
                       dens, f0, onsets, ws_on, ws_f0, ws_sc, ws_norm);
    hipLaunchKernelGGL(k_prep_pw, dim3(GRAINN / TPB), dim3(TPB), 0, stream,
                       slope, pw);
    hipLaunchKernelGGL(k_synth, dim3(NBLK_MAIN), dim3(TPB), 0, stream,
                       (const void*)ws, pw, out, ws_norm);
    hipLaunchKernelGGL(k_norm, dim3(NSAMP / TPB), dim3(TPB), 0, stream,
                       out, ws_norm);
}